// CustomLoss_20675972562957
// MI455X (gfx1250) — compile-verified
//
#include <hip/hip_runtime.h>
#include <math.h>

#define BB 16
#define GG 4000
#define TT 96
#define KK 4
#define PP 500
#define SS 200

typedef float v2f __attribute__((ext_vector_type(2)));
typedef float v8f __attribute__((ext_vector_type(8)));

__device__ __forceinline__ float read_lane_f32(float v, int lane) {
  return __uint_as_float(__builtin_amdgcn_readlane(__float_as_uint(v), lane));
}

// Exact fp32 wave(32)-wide sum using the only f32 WMMA on CDNA5:
// A(16x4): lanes0-15 vgpr0 -> A[m,0]=val_m ; lanes16-31 vgpr0 -> A[m,2]=val_{m+16};
// vgpr1 = 0. B = ones(4x16). D[m,n] = val_m + val_{m+16}.
// Lane<16 sums D[0..7,n]; lane>=16 sums D[8..15,n]; total = s(lane0)+s(lane16).
__device__ __forceinline__ float wave_sum_wmma(float val) {
  v2f a; a[0] = val;  a[1] = 0.0f;
  v2f b; b[0] = 1.0f; b[1] = 1.0f;
  v8f c = {};
  c = __builtin_amdgcn_wmma_f32_16x16x4_f32(false, a, false, b, (short)0, c, false, false);
  float s = c[0] + c[1] + c[2] + c[3] + c[4] + c[5] + c[6] + c[7];
  return read_lane_f32(s, 0) + read_lane_f32(s, 16);
}

__device__ __forceinline__ void block_accum(float val, float scale, float* acc, float* lds) {
  float w = wave_sum_wmma(val);
  const int lane = threadIdx.x & 31;
  const int wid  = threadIdx.x >> 5;
  if (lane == 0) lds[wid] = w;
  __syncthreads();
  if (threadIdx.x == 0) {
    const int nw = ((int)blockDim.x + 31) >> 5;
    float t = 0.0f;
    for (int i = 0; i < nw; ++i) t += lds[i];
    unsafeAtomicAdd(acc, t * scale);
  }
}

__global__ void zero_kernel(float* out) {
  if (threadIdx.x == 0 && blockIdx.x == 0) out[0] = 0.0f;
}

// sum_i t[i]*log(p[i]) + (1-t[i])*log1p(-p[i]) over n4 float4 groups;
// caller passes scale = -1/N.  Requires N % 4 == 0.
__global__ void bce_kernel(const float4* __restrict__ p, const float4* __restrict__ t,
                           int n4, float scale, float* acc) {
  __shared__ float lred[8];
  float partial = 0.0f;
  for (int i = blockIdx.x * blockDim.x + threadIdx.x; i < n4;
       i += gridDim.x * blockDim.x) {
    float4 pv = p[i], tv = t[i];
    partial += tv.x * logf(pv.x) + (1.0f - tv.x) * log1pf(-pv.x);
    partial += tv.y * logf(pv.y) + (1.0f - tv.y) * log1pf(-pv.y);
    partial += tv.z * logf(pv.z) + (1.0f - tv.z) * log1pf(-pv.z);
    partial += tv.w * logf(pv.w) + (1.0f - tv.w) * log1pf(-pv.w);
  }
  block_accum(partial, scale, acc, lred);
}

// sum_i w[(4*i)/wdiv % wmod] * sum(x4[i]) * scale  (w==null -> weight 1)
// Requires N % 4 == 0 and (w != null -> wdiv % 4 == 0) so the weight index
// is uniform within each float4 group.
__global__ void wsum_kernel(const float4* __restrict__ x, const float* __restrict__ w,
                            int wdiv, int wmod, int n4, float scale, float* acc) {
  __shared__ float lred[8];
  float partial = 0.0f;
  for (int i = blockIdx.x * blockDim.x + threadIdx.x; i < n4;
       i += gridDim.x * blockDim.x) {
    float4 v = x[i];
    float s = v.x + v.y + v.z + v.w;
    float wt = (w != nullptr) ? w[((i * 4) / wdiv) % wmod] : 1.0f;
    partial += wt * s;
  }
  block_accum(partial, scale, acc, lred);
}

// sum over (b,g,t): dot4(seg_prod[b,g,t,:], segment_cost[g,:])
__global__ void seg_kernel(const float4* __restrict__ seg, const float4* __restrict__ cost,
                           int nrows, float* acc) {
  __shared__ float lred[8];
  float partial = 0.0f;
  for (int i = blockIdx.x * blockDim.x + threadIdx.x; i < nrows;
       i += gridDim.x * blockDim.x) {
    int g = (i / TT) % GG;
    float4 s = seg[i];
    float4 c = cost[g];
    partial += s.x * c.x + s.y * c.y + s.z * c.z + s.w * c.w;
  }
  block_accum(partial, 1.0f, acc, lred);
}

// One 96-thread block per (b,g) row: startup cost + min-up/down-time violations.
__global__ void row_kernel(const float* __restrict__ rounded,
                           const float* __restrict__ init_commit,
                           const float* __restrict__ su_costs,
                           const int* __restrict__ init_status,
                           const int* __restrict__ min_up,
                           const int* __restrict__ min_dn,
                           float viol_penalty, float* acc) {
  __shared__ float rbuf[TT];
  __shared__ float scn[TT];
  __shared__ float pre[TT + 1];
  __shared__ float lred[8];

  const int row = blockIdx.x;          // 0 .. B*G-1
  const int g   = row % GG;
  const int t   = threadIdx.x;         // 0 .. 95

  const float r = rounded[(size_t)row * TT + t];
  rbuf[t] = r;
  scn[t]  = r;
  __syncthreads();

  // Hillis-Steele inclusive scan over T=96
  for (int off = 1; off < TT; off <<= 1) {
    float v = (t >= off) ? scn[t - off] : 0.0f;
    __syncthreads();
    scn[t] += v;
    __syncthreads();
  }
  pre[t + 1] = scn[t];
  if (t == 0) pre[0] = 0.0f;
  __syncthreads();

  const float prev   = (t == 0) ? init_commit[row] : rbuf[t - 1];
  const float sw_on  = (1.0f - prev) * r;
  const float sw_off = prev * (1.0f - r);

  const int U  = max(min_up[g], 0);
  const int D  = max(min_dn[g], 0);
  const int st = init_status[g];
  const int rem_up = max(U - max(st, 0), 0);
  const int rem_dn = max(D - max(-st, 0), 0);

  float vsum = 0.0f;
  if (t < rem_up) vsum += (1.0f - r);                      // early up
  if (U > 0 && t + U <= TT) {                              // up-window violation
    float wsum = pre[t + U] - pre[t];
    vsum += fmaxf((float)U - wsum, 0.0f) * sw_on;
  }
  if (t < rem_dn) vsum += r;                               // early down
  if (D > 0 && t + D <= TT) {                              // down-window violation
    float wsumd = (float)D - (pre[t + D] - pre[t]);        // sum of (1-r) in window
    vsum += fmaxf((float)D - wsumd, 0.0f) * sw_off;
  }

  float contrib = su_costs[g] * sw_on + viol_penalty * vsum;
  block_accum(contrib, 1.0f, acc, lred);
}

static inline int grid_for(long long n, int block) {
  long long g = (n + block - 1) / block;
  if (g > 4096) g = 4096;
  if (g < 1) g = 1;
  return (int)g;
}

extern "C" void kernel_launch(void* const* d_in, const int* in_sizes, int n_in,
                              void* d_out, int out_size, void* d_ws, size_t ws_size,
                              hipStream_t stream) {
  (void)in_sizes; (void)n_in; (void)out_size; (void)d_ws; (void)ws_size;

  const float* profiled_generation = (const float*)d_in[0];
  const float* charge_rate         = (const float*)d_in[2];
  const float* discharge_rate      = (const float*)d_in[3];
  const float* seg_prod            = (const float*)d_in[4];
  const float* curtailment         = (const float*)d_in[6];
  const float* thermal_on          = (const float*)d_in[7];
  const float* thermal_on_rounded  = (const float*)d_in[8];
  const float* is_charging         = (const float*)d_in[9];
  const float* is_discharging      = (const float*)d_in[10];
  const float* tgt_thermal         = (const float*)d_in[11];
  const float* tgt_charging        = (const float*)d_in[12];
  const float* tgt_discharging     = (const float*)d_in[13];
  const float* initial_commitment  = (const float*)d_in[14];
  const float* segment_cost        = (const float*)d_in[15];
  const float* start_up_costs      = (const float*)d_in[16];
  const float* profiled_units_cost = (const float*)d_in[17];
  const float* charge_costs        = (const float*)d_in[18];
  const float* discharge_costs     = (const float*)d_in[19];
  const int*   initial_status      = (const int*)d_in[20];
  const int*   min_uptimes         = (const int*)d_in[21];
  const int*   min_downtimes       = (const int*)d_in[22];

  float* acc = (float*)d_out;

  const int nBGT = BB * GG * TT;   // 6,144,000
  const int nBST = BB * SS * TT;   // 307,200
  const int nBPT = BB * PP * TT;   // 768,000
  const int nBT  = BB * TT;        // 1,536

  zero_kernel<<<1, 64, 0, stream>>>(acc);

  // Supervised BCE terms (mean -> scale = -1/N), float4 streams
  bce_kernel<<<grid_for(nBGT / 4, 256), 256, 0, stream>>>(
      (const float4*)thermal_on, (const float4*)tgt_thermal,
      nBGT / 4, -1.0f / (float)nBGT, acc);
  bce_kernel<<<grid_for(nBST / 4, 256), 256, 0, stream>>>(
      (const float4*)is_charging, (const float4*)tgt_charging,
      nBST / 4, -1.0f / (float)nBST, acc);
  bce_kernel<<<grid_for(nBST / 4, 256), 256, 0, stream>>>(
      (const float4*)is_discharging, (const float4*)tgt_discharging,
      nBST / 4, -1.0f / (float)nBST, acc);

  // Economic dispatch weighted sums, float4 streams (wdiv=96 is a multiple of 4)
  wsum_kernel<<<grid_for(nBPT / 4, 256), 256, 0, stream>>>(
      (const float4*)profiled_generation, profiled_units_cost, TT, PP, nBPT / 4, 1.0f, acc);
  wsum_kernel<<<grid_for(nBST / 4, 256), 256, 0, stream>>>(
      (const float4*)charge_rate, charge_costs, TT, SS, nBST / 4, 1.0f, acc);
  wsum_kernel<<<grid_for(nBST / 4, 256), 256, 0, stream>>>(
      (const float4*)discharge_rate, discharge_costs, TT, SS, nBST / 4, 1.0f, acc);
  wsum_kernel<<<grid_for(nBT / 4, 256), 256, 0, stream>>>(
      (const float4*)curtailment, nullptr, 1, 1, nBT / 4, 5000.0f, acc);

  // segment_cost * seg_prod (dominant 98 MB stream, float4-coalesced)
  seg_kernel<<<grid_for(nBGT, 256), 256, 0, stream>>>(
      (const float4*)seg_prod, (const float4*)segment_cost, nBGT, acc);

  // Startup costs + min-up/down-time violations: one block per (b,g) row
  row_kernel<<<BB * GG, TT, 0, stream>>>(
      thermal_on_rounded, initial_commitment, start_up_costs,
      initial_status, min_uptimes, min_downtimes, 1000.0f, acc);
}